// HLGNN_89240830476476
// MI455X (gfx1250) — compile-verified
//
#include <hip/hip_runtime.h>
#include <hip/hip_bf16.h>

typedef __attribute__((ext_vector_type(2))) float v2f;
typedef __attribute__((ext_vector_type(8))) float v8f;

#define FDIM 64

// ---------------------------------------------------------------------------
// Generic float4 zero-fill (graph-capture safe replacement for memset)
// ---------------------------------------------------------------------------
__global__ void hlgnn_zero_f4(float4* __restrict__ p, int n4) {
    int i = blockIdx.x * blockDim.x + threadIdx.x;
    if (i < n4) p[i] = make_float4(0.f, 0.f, 0.f, 0.f);
}

// ---------------------------------------------------------------------------
// deg[row[e]] += edge_weight[e]
// ---------------------------------------------------------------------------
__global__ void hlgnn_deg(const int* __restrict__ row, const float* __restrict__ ew,
                          float* __restrict__ deg, int nE) {
    int e = blockIdx.x * blockDim.x + threadIdx.x;
    if (e < nE) atomicAdd(&deg[row[e]], ew[e]);
}

// ---------------------------------------------------------------------------
// in-place deg -> dinv = deg>0 ? rsqrt(deg) : 0
// ---------------------------------------------------------------------------
__global__ void hlgnn_dinv(float* __restrict__ deg, int n) {
    int i = blockIdx.x * blockDim.x + threadIdx.x;
    if (i < n) {
        float d = deg[i];
        deg[i] = (d > 0.f) ? rsqrtf(d) : 0.f;
    }
}

// ---------------------------------------------------------------------------
// w_norm[e] = dinv[row]*ew*dinv[col]
// ---------------------------------------------------------------------------
__global__ void hlgnn_wnorm(const int* __restrict__ row, const int* __restrict__ col,
                            const float* __restrict__ ew, const float* __restrict__ dinv,
                            float* __restrict__ wn, int nE) {
    int e = blockIdx.x * blockDim.x + threadIdx.x;
    if (e < nE) wn[e] = dinv[row[e]] * ew[e] * dinv[col[e]];
}

// ---------------------------------------------------------------------------
// h = x @ W^T + b  via V_WMMA_F32_16X16X4_F32 (full f32 precision).
// One wave computes a 16-row x 64-col output tile: 4 v8f accumulators,
// 16 K-steps of 4. Also writes xk0 = h and hidden = temp[0]*h.
// Tile index is wave-uniform -> EXEC all-ones inside WMMA (ISA requirement).
// ---------------------------------------------------------------------------
__global__ void hlgnn_gemm_wmma(const float* __restrict__ x, const float* __restrict__ W,
                                const float* __restrict__ bias, const float* __restrict__ temp,
                                float* __restrict__ xk, float* __restrict__ hidden,
                                int nTiles) {
    const int wave = blockIdx.x * (blockDim.x >> 5) + (threadIdx.x >> 5);
    if (wave >= nTiles) return;                 // uniform per wave
    const int lane  = threadIdx.x & 31;
    const int lm    = lane & 15;                // row/col within 16-tile
    const int khalf = (lane >> 4) * 2;          // lanes 0-15: K+0/1, 16-31: K+2/3
    const int m0    = wave * 16;

    v8f acc0 = {}, acc1 = {}, acc2 = {}, acc3 = {};

    #pragma unroll
    for (int ks = 0; ks < 16; ++ks) {
        const int k = ks * 4 + khalf;
        // A frag: A[m][k] = x[(m0+lm)*64 + k] ; VGPR0=K+0, VGPR1=K+1
        v2f a;
        a.x = x[(m0 + lm) * FDIM + k];
        a.y = x[(m0 + lm) * FDIM + k + 1];
        // B frag: B[k][n] = W.T[k][n] = W[n*64 + k], n = nt*16 + lm
        v2f b0, b1, b2, b3;
        b0.x = W[(0 * 16 + lm) * FDIM + k];  b0.y = W[(0 * 16 + lm) * FDIM + k + 1];
        b1.x = W[(1 * 16 + lm) * FDIM + k];  b1.y = W[(1 * 16 + lm) * FDIM + k + 1];
        b2.x = W[(2 * 16 + lm) * FDIM + k];  b2.y = W[(2 * 16 + lm) * FDIM + k + 1];
        b3.x = W[(3 * 16 + lm) * FDIM + k];  b3.y = W[(3 * 16 + lm) * FDIM + k + 1];
        acc0 = __builtin_amdgcn_wmma_f32_16x16x4_f32(false, a, false, b0, (short)0, acc0, false, false);
        acc1 = __builtin_amdgcn_wmma_f32_16x16x4_f32(false, a, false, b1, (short)0, acc1, false, false);
        acc2 = __builtin_amdgcn_wmma_f32_16x16x4_f32(false, a, false, b2, (short)0, acc2, false, false);
        acc3 = __builtin_amdgcn_wmma_f32_16x16x4_f32(false, a, false, b3, (short)0, acc3, false, false);
    }

    const float t0 = temp[0];
    const int rbase = m0 + ((lane >> 4) * 8);   // D: lanes 0-15 -> M=vgpr, 16-31 -> M=vgpr+8
    #pragma unroll
    for (int r = 0; r < 8; ++r) {
        const int rowi = rbase + r;
        {   const int c = 0 * 16 + lm;  float hv = acc0[r] + bias[c];
            xk[rowi * FDIM + c] = hv;  hidden[rowi * FDIM + c] = t0 * hv; }
        {   const int c = 1 * 16 + lm;  float hv = acc1[r] + bias[c];
            xk[rowi * FDIM + c] = hv;  hidden[rowi * FDIM + c] = t0 * hv; }
        {   const int c = 2 * 16 + lm;  float hv = acc2[r] + bias[c];
            xk[rowi * FDIM + c] = hv;  hidden[rowi * FDIM + c] = t0 * hv; }
        {   const int c = 3 * 16 + lm;  float hv = acc3[r] + bias[c];
            xk[rowi * FDIM + c] = hv;  hidden[rowi * FDIM + c] = t0 * hv; }
    }
}

// ---------------------------------------------------------------------------
// One propagation hop: xnew[row[e], :] += w_norm[e] * xk[col[e], :]
// Thread layout: 16 threads per edge, 4 features each (float4 gather,
// 4x global_atomic_add_f32 scatter). A wave covers 2 edges -> the 256B
// source rows are read coalesced; col/wn broadcast from L0.
// ---------------------------------------------------------------------------
__global__ void hlgnn_scatter(const int* __restrict__ row, const int* __restrict__ col,
                              const float* __restrict__ wn, const float* __restrict__ xk,
                              float* __restrict__ xnew, int nE) {
    int gid = blockIdx.x * blockDim.x + threadIdx.x;
    int e   = gid >> 4;
    if (e >= nE) return;
    int f0  = (gid & 15) * 4;
    float w = wn[e];
    int c   = col[e];
    int r   = row[e];
    float4 v = *(const float4*)(xk + (size_t)c * FDIM + f0);
    float* dst = xnew + (size_t)r * FDIM + f0;
    atomicAdd(dst + 0, w * v.x);
    atomicAdd(dst + 1, w * v.y);
    atomicAdd(dst + 2, w * v.z);
    atomicAdd(dst + 3, w * v.w);
}

// ---------------------------------------------------------------------------
// hidden += temp[k] * xk   (float4, fully coalesced)
// ---------------------------------------------------------------------------
__global__ void hlgnn_axpy(float4* __restrict__ hidden, const float4* __restrict__ xk,
                           const float* __restrict__ temp, int k, int n4) {
    int i = blockIdx.x * blockDim.x + threadIdx.x;
    if (i < n4) {
        float g = temp[k];       // scalar (uniform) load
        float4 h = hidden[i];
        float4 x = xk[i];
        h.x += g * x.x;  h.y += g * x.y;  h.z += g * x.z;  h.w += g * x.w;
        hidden[i] = h;
    }
}

// ---------------------------------------------------------------------------
// Host-side orchestration
// ---------------------------------------------------------------------------
extern "C" void kernel_launch(void* const* d_in, const int* in_sizes, int n_in,
                              void* d_out, int out_size, void* d_ws, size_t ws_size,
                              hipStream_t stream) {
    const float* x    = (const float*)d_in[0];
    const int*   ei   = (const int*)  d_in[1];
    const float* ew   = (const float*)d_in[2];
    const float* W    = (const float*)d_in[3];
    const float* bias = (const float*)d_in[4];
    const float* temp = (const float*)d_in[5];

    const int N = in_sizes[0] / FDIM;
    const int E = in_sizes[2];
    const int K = in_sizes[5] - 1;

    const int* rowIdx = ei;        // edge_index[0]
    const int* colIdx = ei + E;    // edge_index[1]

    // workspace layout (256B aligned): dinv | w_norm | xkA | xkB
    auto align256 = [](size_t s) { return (s + 255) & ~(size_t)255; };
    char* ws = (char*)d_ws;
    size_t off = 0;
    float* dinv  = (float*)(ws + off);  off += align256((size_t)N * sizeof(float));
    float* wnorm = (float*)(ws + off);  off += align256((size_t)E * sizeof(float));
    float* xkA   = (float*)(ws + off);  off += align256((size_t)N * FDIM * sizeof(float));
    float* xkB   = (float*)(ws + off);
    float* hidden = (float*)d_out;

    const int nNF  = N * FDIM;
    const int nNF4 = nNF / 4;
    const int B = 256;

    // 1) degree: zero then scatter-add
    hlgnn_zero_f4<<<(N / 4 + B - 1) / B, B, 0, stream>>>((float4*)dinv, N / 4);
    hlgnn_deg<<<(E + B - 1) / B, B, 0, stream>>>(rowIdx, ew, dinv, E);
    // 2) dinv in place
    hlgnn_dinv<<<(N + B - 1) / B, B, 0, stream>>>(dinv, N);
    // 3) edge normalization
    hlgnn_wnorm<<<(E + B - 1) / B, B, 0, stream>>>(rowIdx, colIdx, ew, dinv, wnorm, E);
    // 4) h = x@W.T + b via WMMA; xkA = h; hidden = temp[0]*h
    const int nTiles = N / 16;                       // N = 100000 is a multiple of 16
    const int wavesPerBlk = B / 32;
    hlgnn_gemm_wmma<<<(nTiles + wavesPerBlk - 1) / wavesPerBlk, B, 0, stream>>>(
        x, W, bias, temp, xkA, hidden, nTiles);

    // 5) K hops of propagation with weighted accumulation
    float* xk = xkA;
    float* xn = xkB;
    const int scatterThreads = E * 16;
    for (int k = 1; k <= K; ++k) {
        hlgnn_zero_f4<<<(nNF4 + B - 1) / B, B, 0, stream>>>((float4*)xn, nNF4);
        hlgnn_scatter<<<(scatterThreads + B - 1) / B, B, 0, stream>>>(
            rowIdx, colIdx, wnorm, xk, xn, E);
        hlgnn_axpy<<<(nNF4 + B - 1) / B, B, 0, stream>>>(
            (float4*)hidden, (const float4*)xn, temp, k, nNF4);
        float* t = xk; xk = xn; xn = t;
    }
}